// DETRLoss_59279138619473
// MI455X (gfx1250) — compile-verified
//
#include <hip/hip_runtime.h>
#include <stdint.h>

// ---------------- problem constants (from reference) ----------------
#define NT      256      // 8 wave32s per block
#define NWAVES  8
#define SLOTS   300      // S
#define SP      304      // S padded to 19 WMMA tiles
#define GMAX    100      // G
#define GP      112      // G padded to 7 WMMA tiles
#define STILES  19
#define CMS     304      // cost-matrix row stride (floats)

typedef __attribute__((ext_vector_type(2))) float v2f;
typedef __attribute__((ext_vector_type(8))) float v8f;

// ---------------- LDS layout (dynamic extern shared) ----------------
// float region
#define OFF_CM    0                       // [GP][CMS] transposed cost C[g][s]
#define OFF_PS    (OFF_CM + GP * CMS)     // strokes [SP][10] (16B aligned)
#define OFF_CLS   (OFF_PS + SP * 10)      // -CLASS_W*probs padded K=4 [SP][4]
#define OFF_LSE   (OFF_CLS + SP * 4)      // logsumexp per slot [SP]
#define OFF_VT    (OFF_LSE + SP)          // compacted valid targets [GP][10]
#define OFF_U     (OFF_VT + GP * 10)      // LAP dual u [GP]
#define OFF_V     (OFF_U + GP)            // LAP dual v [SP]
#define OFF_SHORT (OFF_V + SP)            // shortest [SP]
#define OFF_REDV  (OFF_SHORT + SP)        // reduction values [NT]
#define OFF_INT   (OFF_REDV + NT)         // int region begins (float index)
// int region (indices relative to OFF_INT)
#define IOFF_GTCLS 0
#define IOFF_VALID (IOFF_GTCLS + GP)
#define IOFF_PATH  (IOFF_VALID + GP)
#define IOFF_SC    (IOFF_PATH + SP)
#define IOFF_R4C   (IOFF_SC + SP)
#define IOFF_C4R   (IOFF_R4C + SP)
#define IOFF_SR    (IOFF_C4R + GP)
#define IOFF_TC    (IOFF_SR + GP)
#define IOFF_REDI  (IOFF_TC + SP)
#define IOFF_SCAL  (IOFF_REDI + NT)       // 8 ints of scalars
#define IOFF_END   (IOFF_SCAL + 8)
#define SMEM_FLOATS (OFF_INT + IOFF_END + 4)   // +4 floats for fscal
#define SMEM_BYTES  (SMEM_FLOATS * 4)

// ---------------- CDNA5 async global->LDS (guarded) ----------------
#if defined(__HIP_DEVICE_COMPILE__) && __has_builtin(__builtin_amdgcn_global_load_async_to_lds_b32)
#define ASYNC_LDS 1
#else
#define ASYNC_LDS 0
#endif

#if ASYNC_LDS && __has_builtin(__builtin_amdgcn_global_load_async_to_lds_b128)
#define ASYNC_LDS_B128 1
#else
#define ASYNC_LDS_B128 0
#endif

#if ASYNC_LDS
typedef __attribute__((address_space(1))) int as1_int;
typedef __attribute__((address_space(3))) int as3_int;
typedef __attribute__((__vector_size__(16))) int v4i_t;
typedef __attribute__((address_space(1))) v4i_t as1_v4i;
typedef __attribute__((address_space(3))) v4i_t as3_v4i;
#endif

__device__ __forceinline__ void async_copy_b32(const float* g, float* l) {
#if ASYNC_LDS
  __builtin_amdgcn_global_load_async_to_lds_b32(
      (as1_int*)(uintptr_t)g, (as3_int*)(uint32_t)(uintptr_t)l, 0, 0);
#else
  *l = *g;
#endif
}

__device__ __forceinline__ void async_copy_b128(const float* g, float* l) {
#if ASYNC_LDS_B128
  __builtin_amdgcn_global_load_async_to_lds_b128(
      (as1_v4i*)(uintptr_t)g, (as3_v4i*)(uint32_t)(uintptr_t)l, 0, 0);
#else
  async_copy_b32(g + 0, l + 0);
  async_copy_b32(g + 1, l + 1);
  async_copy_b32(g + 2, l + 2);
  async_copy_b32(g + 3, l + 3);
#endif
}

__device__ __forceinline__ void async_wait_all() {
#if ASYNC_LDS
#if __has_builtin(__builtin_amdgcn_s_wait_asynccnt)
  __builtin_amdgcn_s_wait_asynccnt(0);
#else
  asm volatile("s_wait_asynccnt 0" ::: "memory");
#endif
#endif
}

// ================= per-batch: cost (WMMA) + LAP + partial losses =================
__global__ __launch_bounds__(NT, 1)
void detr_match_kernel(const float* __restrict__ strokes,   // [B,300,10]
                       const float* __restrict__ logits,    // [B,300,3]
                       const float* __restrict__ tparams,   // [B,100,10]
                       const int*   __restrict__ tlabels,   // [B,100]
                       float* __restrict__ part)            // [B,6] partial sums
{
  extern __shared__ float smem[];
  float* CM     = smem + OFF_CM;
  float* psL    = smem + OFF_PS;
  float* clsL   = smem + OFF_CLS;
  float* lseL   = smem + OFF_LSE;
  float* vtL    = smem + OFF_VT;
  float* uL     = smem + OFF_U;
  float* vL     = smem + OFF_V;
  float* shortL = smem + OFF_SHORT;
  float* redV   = smem + OFF_REDV;
  int*   ip     = (int*)(smem + OFF_INT);
  int*   gtclsL = ip + IOFF_GTCLS;
  int*   validL = ip + IOFF_VALID;
  int*   pathL  = ip + IOFF_PATH;
  int*   SCL    = ip + IOFF_SC;
  int*   r4cL   = ip + IOFF_R4C;
  int*   c4rL   = ip + IOFF_C4R;
  int*   SRL    = ip + IOFF_SR;
  int*   tcL    = ip + IOFF_TC;
  int*   redI   = ip + IOFF_REDI;
  int*   scal   = ip + IOFF_SCAL;          // [0]=nv [1]=sink [2]=i [3]=SRcount
  float* fscal  = (float*)(ip + IOFF_END); // [0]=minVal

  const int b   = blockIdx.x;
  const int tid = threadIdx.x;

  // ---- phase 0: compact valid targets (labels > 0) ----
  if (tid == 0) {
    int nv = 0;
    for (int t = 0; t < GMAX; ++t) {
      int lab = tlabels[(size_t)b * GMAX + t];
      if (lab > 0) { validL[nv] = t; gtclsL[nv] = lab; ++nv; }
    }
    scal[0] = nv;
  }
  __syncthreads();
  const int nv = scal[0];
  for (int g = tid; g < GP; g += NT)
    if (g >= nv) { gtclsL[g] = 0; validL[g] = 0; }

  // ---- phase 1: stage strokes into LDS (wide async path on CDNA5) ----
  const float* sb = strokes + (size_t)b * SLOTS * 10;
  for (int i = tid; i < (SLOTS * 10) / 4; i += NT)      // 750 x b128, both sides 16B aligned
    async_copy_b128(sb + i * 4, psL + i * 4);
  async_wait_all();
  for (int i = SLOTS * 10 + tid; i < SP * 10; i += NT) psL[i] = 0.f;

  for (int idx = tid; idx < GP * 10; idx += NT) {
    int g = idx / 10, k = idx - g * 10;
    vtL[idx] = (g < nv)
        ? tparams[((size_t)b * GMAX + validL[g]) * 10 + k] : 0.f;
  }
  const float* lgb = logits + (size_t)b * SLOTS * 3;
  for (int s = tid; s < SP; s += NT) {
    if (s < SLOTS) {
      float x0 = lgb[s * 3 + 0], x1 = lgb[s * 3 + 1], x2 = lgb[s * 3 + 2];
      float mx = fmaxf(x0, fmaxf(x1, x2));
      float e0 = expf(x0 - mx), e1 = expf(x1 - mx), e2 = expf(x2 - mx);
      float sum = e0 + e1 + e2;
      lseL[s] = mx + logf(sum);
      float inv = 1.0f / sum;                 // CLASS_W == 1
      clsL[s * 4 + 0] = -(e0 * inv);
      clsL[s * 4 + 1] = -(e1 * inv);
      clsL[s * 4 + 2] = -(e2 * inv);
      clsL[s * 4 + 3] = 0.f;
    } else {
      lseL[s] = 0.f;
      clsL[s * 4 + 0] = clsL[s * 4 + 1] = clsL[s * 4 + 2] = clsL[s * 4 + 3] = 0.f;
    }
  }
  __syncthreads();

  // ---- phase 2: fused cost tiles. D = (-probs)x(onehot) + weightedL1  via WMMA ----
  if (nv > 0) {
    const int gtiles = (nv + 15) >> 4;
    const int tiles  = gtiles * STILES;
    const int wave = tid >> 5, lane = tid & 31, l16 = lane & 15, hi = lane >> 4;
    for (int t = wave; t < tiles; t += NWAVES) {       // wave-uniform => EXEC all 1s
      const int gt = t / STILES, st = t - gt * STILES;
      const int g0 = gt << 4, s0 = st << 4;
      const int gcol = g0 + l16;
      // A operand: A[M][k] = clsL[s0+M][k]  (16x4 f32 layout per ISA)
      const int arow = s0 + l16;
      v2f a; a.x = clsL[arow * 4 + 2 * hi]; a.y = clsL[arow * 4 + 2 * hi + 1];
      // B operand: onehot(class of target column)
      const int gc = gtclsL[gcol];
      v2f bm; bm.x = (gc == 2 * hi) ? 1.f : 0.f; bm.y = (gc == 2 * hi + 1) ? 1.f : 0.f;
      // C accumulator: weighted L1  (k0,1: 5+2=7; k2..7: 5; k8,9: 2)
      float tv[10];
#pragma unroll
      for (int k = 0; k < 10; ++k) tv[k] = vtL[gcol * 10 + k];
      v8f c;
#pragma unroll
      for (int r = 0; r < 8; ++r) {
        const int srow = s0 + r + 8 * hi;              // C/D layout: M = r + 8*hi
        const float* p = psL + srow * 10;
        float l = 0.f;
#pragma unroll
        for (int k = 0; k < 10; ++k) {
          const float wk = (k < 2) ? 7.f : ((k < 8) ? 5.f : 2.f);
          l += wk * fabsf(p[k] - tv[k]);
        }
        c[r] = (srow < SLOTS) ? l : 1e30f;             // padded slots never win
      }
      v8f d = __builtin_amdgcn_wmma_f32_16x16x4_f32(
          false, a, false, bm, (short)0, c, false, false);
#pragma unroll
      for (int r = 0; r < 8; ++r) {
        const int srow = s0 + r + 8 * hi;
        CM[(size_t)gcol * CMS + srow] = d[r];          // store transposed C[g][s]
      }
    }
  }
  __syncthreads();

  // ---- phase 3: Jonker-Volgenant LAP (rows = targets nv, cols = slots 300) ----
  if (nv > 0) {
    for (int j = tid; j < SP; j += NT) { vL[j] = 0.f; r4cL[j] = -1; }
    for (int i = tid; i < GP; i += NT) { uL[i] = 0.f; c4rL[i] = -1; }
    __syncthreads();
    for (int cur = 0; cur < nv; ++cur) {
      for (int j = tid; j < SP; j += NT) {
        shortL[j] = 3.0e38f; pathL[j] = -1; SCL[j] = (j >= SLOTS) ? 1 : 0;
      }
      if (tid == 0) { scal[1] = -1; scal[2] = cur; scal[3] = 0; fscal[0] = 0.f; }
      __syncthreads();
      while (true) {
        const int   irow   = scal[2];
        const float minVal = fscal[0];
        if (tid == 0) { SRL[scal[3]] = irow; scal[3] = scal[3] + 1; }
        const float ui = uL[irow];
        const float* Ci = CM + (size_t)irow * CMS;
        for (int j = tid; j < SLOTS; j += NT) {
          if (!SCL[j]) {
            float dj = minVal + Ci[j] - ui - vL[j];
            if (dj < shortL[j]) { shortL[j] = dj; pathL[j] = irow; }
          }
        }
        __syncthreads();
        // argmin over masked shortest (numpy tie-break: lowest index)
        float bv = 3.0e38f; int bj = SP;
        for (int j = tid; j < SP; j += NT) {
          float val = SCL[j] ? 3.0e38f : shortL[j];
          if (val < bv) { bv = val; bj = j; }
        }
        redV[tid] = bv; redI[tid] = bj;
        __syncthreads();
        for (int off = NT / 2; off > 0; off >>= 1) {
          if (tid < off) {
            float ov = redV[tid + off]; int oj = redI[tid + off];
            if (ov < redV[tid] || (ov == redV[tid] && oj < redI[tid])) {
              redV[tid] = ov; redI[tid] = oj;
            }
          }
          __syncthreads();
        }
        if (tid == 0) {
          int j = redI[0];
          fscal[0] = redV[0];
          SCL[j] = 1;
          int rc = r4cL[j];
          if (rc < 0) scal[1] = j; else scal[2] = rc;
        }
        __syncthreads();
        if (scal[1] >= 0) break;
      }
      // dual updates
      const float minVal = fscal[0];
      const int   srn    = scal[3];
      if (tid == 0) uL[cur] += minVal;
      for (int k = tid; k < srn; k += NT) {
        int r = SRL[k];
        if (r != cur) uL[r] += minVal - shortL[c4rL[r]];
      }
      for (int j = tid; j < SLOTS; j += NT)
        if (SCL[j]) vL[j] -= minVal - shortL[j];
      __syncthreads();
      // augment (serial, thread 0)
      if (tid == 0) {
        int j = scal[1];
        while (true) {
          int i2 = pathL[j];
          r4cL[j] = i2;
          int nxt = c4rL[i2];
          c4rL[i2] = j;
          if (i2 == cur) break;
          j = nxt;
        }
      }
      __syncthreads();
    }
  }

  // ---- phase 4: losses ----
  for (int s = tid; s < SP; s += NT) tcL[s] = 0;
  __syncthreads();
  for (int g = tid; g < nv; g += NT) tcL[c4rL[g]] = gtclsL[g];
  __syncthreads();

  float ce_num = 0.f, ce_den = 0.f;
  for (int s = tid; s < SLOTS; s += NT) {
    int c = tcL[s];
    float w = (c == 0) ? 0.1f : 1.0f;                  // PEN_W
    float nll = lseL[s] - lgb[s * 3 + c];              // -log_softmax[c]
    ce_num += w * nll; ce_den += w;
  }
  float l1s = 0.f, sl1s = 0.f, wds = 0.f;
  for (int g = tid; g < nv; g += NT) {
    int r = c4rL[g];
    const float* p = psL + r * 10;
    const float* q = vtL + g * 10;
#pragma unroll
    for (int k = 0; k < 8; ++k) {
      float ad = fabsf(p[k] - q[k]);
      l1s  += ad;
      sl1s += (ad < 0.1f) ? (5.0f * ad * ad) : (ad - 0.05f);  // SmoothL1, beta=0.1
    }
    wds += fabsf(p[8] - q[8]) + fabsf(p[9] - q[9]);
  }
  float vals[6] = { ce_num, ce_den, l1s, sl1s, wds, (tid == 0) ? (float)nv : 0.f };
  for (int q = 0; q < 6; ++q) {
    __syncthreads();
    redV[tid] = vals[q];
    __syncthreads();
    for (int off = NT / 2; off > 0; off >>= 1) {
      if (tid < off) redV[tid] += redV[tid + off];
      __syncthreads();
    }
    if (tid == 0) part[(size_t)b * 6 + q] = redV[0];
  }
}

// ================= deterministic final combine =================
__global__ void detr_final_kernel(const float* __restrict__ part, int B,
                                  float* __restrict__ out) {
  float a0 = 0.f, a1 = 0.f, a2 = 0.f, a3 = 0.f, a4 = 0.f, a5 = 0.f;
  for (int b = 0; b < B; ++b) {
    a0 += part[b * 6 + 0]; a1 += part[b * 6 + 1]; a2 += part[b * 6 + 2];
    a3 += part[b * 6 + 3]; a4 += part[b * 6 + 4]; a5 += part[b * 6 + 5];
  }
  float ce    = a0 / a1;
  float denom = fmaxf(a5, 1.0f);
  out[0] = ce + 5.0f * ((a2 + a3) / denom) + 2.0f * (a4 / denom);
}

extern "C" void kernel_launch(void* const* d_in, const int* in_sizes, int n_in,
                              void* d_out, int out_size, void* d_ws, size_t ws_size,
                              hipStream_t stream) {
  (void)n_in; (void)out_size; (void)ws_size;
  const float* strokes = (const float*)d_in[0];
  const float* logits  = (const float*)d_in[1];
  const float* tparams = (const float*)d_in[2];
  const int*   tlabels = (const int*)d_in[3];
  float* part = (float*)d_ws;                     // [B,6] partial sums
  float* out  = (float*)d_out;

  const int B = in_sizes[0] / (SLOTS * 10);       // 128

  (void)hipFuncSetAttribute((const void*)detr_match_kernel,
                            hipFuncAttributeMaxDynamicSharedMemorySize,
                            (int)SMEM_BYTES);
  detr_match_kernel<<<B, NT, SMEM_BYTES, stream>>>(strokes, logits, tparams,
                                                   tlabels, part);
  detr_final_kernel<<<1, 1, 0, stream>>>(part, B, out);
}